// taGCN_16286515986691
// MI455X (gfx1250) — compile-verified
//
#include <hip/hip_runtime.h>
#include <math.h>

typedef float v2f __attribute__((ext_vector_type(2)));
typedef float v8f __attribute__((ext_vector_type(8)));

#define NN   50000
#define EE   800000
#define CDIM 128
#define OUTC 40
#define MROWS 80   // rows per GEMM block: 50000 = 625 * 80 (no M masking)

// ---------------------------------------------------------------- utilities
__global__ void fill_zero_k(float* __restrict__ p, int n) {
    int i = blockIdx.x * blockDim.x + threadIdx.x;
    if (i < n) p[i] = 0.0f;
}

__global__ void fill_bias_k(float* __restrict__ c, const float* __restrict__ b,
                            int ncols, int n) {
    int i = blockIdx.x * blockDim.x + threadIdx.x;
    if (i < n) c[i] = b[i % ncols];
}

__global__ void relu_k(float* __restrict__ p, int n) {
    int i = blockIdx.x * blockDim.x + threadIdx.x;
    if (i < n) p[i] = fmaxf(p[i], 0.0f);
}

// ------------------------------------------------------- graph normalization
__global__ void deg_count_k(const int* __restrict__ dst, float* __restrict__ deg, int E) {
    int e = blockIdx.x * blockDim.x + threadIdx.x;
    if (e < E) atomicAdd(&deg[dst[e]], 1.0f);
}

__global__ void deg_to_dis_k(float* __restrict__ deg, int n) {
    int i = blockIdx.x * blockDim.x + threadIdx.x;
    if (i < n) {
        float d = deg[i];
        deg[i] = (d > 0.0f) ? rsqrtf(fmaxf(d, 1.0f)) : 0.0f;
    }
}

__global__ void edge_norm_k(const int* __restrict__ src, const int* __restrict__ dst,
                            const float* __restrict__ dis, float* __restrict__ nrm, int E) {
    int e = blockIdx.x * blockDim.x + threadIdx.x;
    if (e < E) nrm[e] = dis[src[e]] * dis[dst[e]];
}

// ------------------------------------------------- sparse propagation (SpMM)
// One wave per edge: 32 lanes each move a float4 (coalesced 512B row).
// Feature matrix (25.6 MB) is L2-resident on MI455X (192 MB L2), so the
// random gather + scatter-atomics run at L2 speed, not HBM.
__global__ void spmm_k(const int* __restrict__ src, const int* __restrict__ dst,
                       const float* __restrict__ nrm,
                       const float* __restrict__ h, float* __restrict__ out, int E) {
    int e    = (int)((blockIdx.x * (unsigned)blockDim.x + threadIdx.x) >> 5);
    int lane = threadIdx.x & 31;
    if (e >= E) return;
    int   s = src[e];
    int   d = dst[e];
    float w = nrm[e];
    const float4* hp = (const float4*)(h + (size_t)s * CDIM);
    float4 v = hp[lane];
    float* op = out + (size_t)d * CDIM + lane * 4;
    atomicAdd(op + 0, w * v.x);
    atomicAdd(op + 1, w * v.y);
    atomicAdd(op + 2, w * v.z);
    atomicAdd(op + 3, w * v.w);
}

// ------------------------------------------------------ fp32 WMMA GEMM-acc
// C[M x NC] += A[M x 128] @ W[128 x NC], accumulating in place (C pre-seeded
// with bias). One block covers an 80-row M panel (5 M-tiles); NT = ceil(NC/16)
// waves each own one 16-col N tile and 5 accumulators. A panel and the
// (zero-padded) W slice are staged in LDS, so the K loop is pure
// ds_load_b64 + v_wmma_f32_16x16x4_f32 with EXEC untouched (no divergent
// guards inside the loop). B staging is pair-interleaved so each lane's two
// K-elements are adjacent -> single ds_load_b64 per B fragment.
template <int NC>
__global__ void gemm_wmma_acc_k(const float* __restrict__ A,
                                const float* __restrict__ W,
                                float* __restrict__ Cm) {
    constexpr int NT   = (NC + 15) / 16;   // N tiles (waves) per block
    constexpr int NTC  = NT * 16;          // padded column count
    constexpr int MTB  = MROWS / 16;       // 5 M tiles per block
    constexpr int NTHR = NT * 32;

    __shared__ float As[MROWS * CDIM];     // row-major panel (contiguous copy)
    __shared__ float Bs[CDIM * NTC];       // [(k>>1)*NTC + n]*2 + (k&1)

    const int M0  = blockIdx.x * MROWS;
    const int tid = threadIdx.x;

    const float* Ab = A + (size_t)M0 * CDIM;
    for (int i = tid; i < MROWS * CDIM; i += NTHR) As[i] = Ab[i];
    for (int i = tid; i < CDIM * NTC; i += NTHR) {
        int k = i / NTC, n = i % NTC;
        float v = (n < NC) ? W[(size_t)k * NC + n] : 0.0f;
        Bs[((k >> 1) * NTC + n) * 2 + (k & 1)] = v;
    }
    __syncthreads();

    const int wave  = tid >> 5;
    const int lane  = tid & 31;
    const int l15   = lane & 15;
    const int hi    = lane >> 4;           // lane half selects K pair {0,1} vs {2,3}
    const int nloc  = wave * 16 + l15;     // output column
    const bool nvalid = (nloc < NC);

    // C fragment layout (ISA): VGPR v -> M = v + 8*hi, N = lane&15
    v8f acc[MTB];
#pragma unroll
    for (int t = 0; t < MTB; ++t) {
#pragma unroll
        for (int v = 0; v < 8; ++v) {
            int row = M0 + t * 16 + v + hi * 8;
            acc[t][v] = nvalid ? Cm[(size_t)row * NC + nloc] : 0.0f;
        }
    }

#pragma unroll 4
    for (int k = 0; k < CDIM; k += 4) {
        v2f b;
        const float* bp = &Bs[(((k >> 1) + hi) * NTC + nloc) * 2];
        b[0] = bp[0];
        b[1] = bp[1];
#pragma unroll
        for (int t = 0; t < MTB; ++t) {
            v2f a;
            const float* ap = &As[(t * 16 + l15) * CDIM + k + 2 * hi];
            a[0] = ap[0];
            a[1] = ap[1];
            acc[t] = __builtin_amdgcn_wmma_f32_16x16x4_f32(
                    /*neg_a=*/false, a, /*neg_b=*/false, b,
                    /*c_mod=*/(short)0, acc[t], /*reuse_a=*/false, /*reuse_b=*/false);
        }
    }

#pragma unroll
    for (int t = 0; t < MTB; ++t) {
#pragma unroll
        for (int v = 0; v < 8; ++v) {
            int row = M0 + t * 16 + v + hi * 8;
            if (nvalid) Cm[(size_t)row * NC + nloc] = acc[t][v];
        }
    }
}

// -------------------------------------------------------------- log_softmax
__global__ void log_softmax_k(float* __restrict__ out, int M) {
    int r = blockIdx.x * blockDim.x + threadIdx.x;
    if (r >= M) return;
    float* p = out + (size_t)r * OUTC;
    float mx = -INFINITY;
#pragma unroll
    for (int j = 0; j < OUTC; ++j) mx = fmaxf(mx, p[j]);
    float s = 0.0f;
#pragma unroll
    for (int j = 0; j < OUTC; ++j) s += __expf(p[j] - mx);
    float lse = mx + __logf(s);
#pragma unroll
    for (int j = 0; j < OUTC; ++j) p[j] = p[j] - lse;
}

// ------------------------------------------------------------------- driver
extern "C" void kernel_launch(void* const* d_in, const int* in_sizes, int n_in,
                              void* d_out, int out_size, void* d_ws, size_t ws_size,
                              hipStream_t stream) {
    const float* x   = (const float*)d_in[0];
    const int*   ei  = (const int*)  d_in[1];
    const float* W1  = (const float*)d_in[2];
    const float* b1  = (const float*)d_in[3];
    const float* W2  = (const float*)d_in[4];
    const float* b2  = (const float*)d_in[5];
    float*       out = (float*)d_out;

    const int* src = ei;        // edge_index[0, :]
    const int* dst = ei + EE;   // edge_index[1, :]

    // workspace carve-up (all 16B-aligned slices)
    char*  ws   = (char*)d_ws;
    float* dis  = (float*)ws;  ws += sizeof(float) * NN;          // deg -> dis
    float* nrm  = (float*)ws;  ws += sizeof(float) * EE;          // per-edge norm
    float* bufA = (float*)ws;  ws += sizeof(float) * (size_t)NN * CDIM;
    float* bufB = (float*)ws;  ws += sizeof(float) * (size_t)NN * CDIM;
    float* acc1 = (float*)ws;  ws += sizeof(float) * (size_t)NN * CDIM;

    const int T  = 256;
    const int MB = NN / MROWS;          // 625 GEMM blocks
    auto gr = [](int n, int t) { return (n + t - 1) / t; };

    // ---- symmetric normalization
    fill_zero_k<<<gr(NN, T), T, 0, stream>>>(dis, NN);
    deg_count_k<<<gr(EE, T), T, 0, stream>>>(dst, dis, EE);
    deg_to_dis_k<<<gr(NN, T), T, 0, stream>>>(dis, NN);
    edge_norm_k<<<gr(EE, T), T, 0, stream>>>(src, dst, dis, nrm, EE);

    const int SPMM_GRID = gr(EE, T / 32);  // 8 edges per 256-thread block

    // ---- layer 1: acc1 = b1 + sum_c xs_c @ W1[c*128:(c+1)*128, :]
    fill_bias_k<<<gr(NN * CDIM, T), T, 0, stream>>>(acc1, b1, CDIM, NN * CDIM);
    gemm_wmma_acc_k<CDIM><<<MB, 8 * 32, 0, stream>>>(x, W1 + 0 * CDIM * CDIM, acc1);

    fill_zero_k<<<gr(NN * CDIM, T), T, 0, stream>>>(bufA, NN * CDIM);
    spmm_k<<<SPMM_GRID, T, 0, stream>>>(src, dst, nrm, x, bufA, EE);
    gemm_wmma_acc_k<CDIM><<<MB, 8 * 32, 0, stream>>>(bufA, W1 + 1 * CDIM * CDIM, acc1);

    fill_zero_k<<<gr(NN * CDIM, T), T, 0, stream>>>(bufB, NN * CDIM);
    spmm_k<<<SPMM_GRID, T, 0, stream>>>(src, dst, nrm, bufA, bufB, EE);
    gemm_wmma_acc_k<CDIM><<<MB, 8 * 32, 0, stream>>>(bufB, W1 + 2 * CDIM * CDIM, acc1);

    fill_zero_k<<<gr(NN * CDIM, T), T, 0, stream>>>(bufA, NN * CDIM);
    spmm_k<<<SPMM_GRID, T, 0, stream>>>(src, dst, nrm, bufB, bufA, EE);
    gemm_wmma_acc_k<CDIM><<<MB, 8 * 32, 0, stream>>>(bufA, W1 + 3 * CDIM * CDIM, acc1);

    relu_k<<<gr(NN * CDIM, T), T, 0, stream>>>(acc1, NN * CDIM);

    // ---- layer 2: out = b2 + sum_c hs_c @ W2[c*128:(c+1)*128, :]
    fill_bias_k<<<gr(NN * OUTC, T), T, 0, stream>>>(out, b2, OUTC, NN * OUTC);
    gemm_wmma_acc_k<OUTC><<<MB, 3 * 32, 0, stream>>>(acc1, W2 + 0 * CDIM * OUTC, out);

    fill_zero_k<<<gr(NN * CDIM, T), T, 0, stream>>>(bufB, NN * CDIM);
    spmm_k<<<SPMM_GRID, T, 0, stream>>>(src, dst, nrm, acc1, bufB, EE);
    gemm_wmma_acc_k<OUTC><<<MB, 3 * 32, 0, stream>>>(bufB, W2 + 1 * CDIM * OUTC, out);

    fill_zero_k<<<gr(NN * CDIM, T), T, 0, stream>>>(bufA, NN * CDIM);
    spmm_k<<<SPMM_GRID, T, 0, stream>>>(src, dst, nrm, bufB, bufA, EE);
    gemm_wmma_acc_k<OUTC><<<MB, 3 * 32, 0, stream>>>(bufA, W2 + 2 * CDIM * OUTC, out);

    fill_zero_k<<<gr(NN * CDIM, T), T, 0, stream>>>(bufB, NN * CDIM);
    spmm_k<<<SPMM_GRID, T, 0, stream>>>(src, dst, nrm, bufA, bufB, EE);
    gemm_wmma_acc_k<OUTC><<<MB, 3 * 32, 0, stream>>>(bufB, W2 + 3 * CDIM * OUTC, out);

    // ---- log_softmax over 40 classes, in place on d_out
    log_softmax_k<<<gr(NN, T), T, 0, stream>>>(out, NN);
}